// SigLipConceptLoss_7894149890369
// MI455X (gfx1250) — compile-verified
//
#include <hip/hip_runtime.h>

// ---------------------------------------------------------------------------
// Span mean-pooling for MI455X (gfx1250, wave32).
//   pooled[b,s,:] = mean over len rows of embeddings[b, start..start+len-1, :]
//   mask[b,s]     = (s < span_nums[b])
// One workgroup (256 threads = 8 waves) per span.
// Stage rows into LDS with GLOBAL_LOAD_ASYNC_TO_LDS_B128 (ASYNCcnt path),
// reduce with chained V_WMMA_F32_16X16X4_F32 (A = broadcast weights,
// B = gathered rows from LDS, f32 accumulate).
// ---------------------------------------------------------------------------

typedef float v2f __attribute__((ext_vector_type(2)));
typedef float v8f __attribute__((ext_vector_type(8)));
typedef int   v4i __attribute__((vector_size(16)));   // matches builtin param

#define AS1 __attribute__((address_space(1)))
#define AS3 __attribute__((address_space(3)))

#if defined(__has_builtin)
#  if __has_builtin(__builtin_amdgcn_global_load_async_to_lds_b128)
#    define HAVE_ASYNC_B128 1
#  endif
#  if __has_builtin(__builtin_amdgcn_s_wait_asynccnt)
#    define HAVE_WAIT_ASYNC 1
#  endif
#endif

static constexpr int Bc = 64;
static constexpr int Lc = 2048;
static constexpr int Dc = 768;
static constexpr int Sc = 32;
static constexpr int MAX_ROWS = 16;
// 768 + 16 dwords pad: rows 2 apart land 32 banks apart -> the two half-waves'
// 16-lane B-operand reads touch disjoint bank ranges.
static constexpr int LDS_STRIDE = 784;
static constexpr int ROW_CHUNKS16 = (Dc * 4) / 16;   // 192 x 16B per row

__device__ __forceinline__ void async_copy_b128(const float* gsrc,
                                                float* lds_generic,
                                                unsigned lds_byte_off) {
#if defined(HAVE_ASYNC_B128)
  __builtin_amdgcn_global_load_async_to_lds_b128(
      (AS1 v4i*)gsrc,
      (AS3 v4i*)((AS3 char*)lds_generic + lds_byte_off),
      0, 0);
#else
  // Fallback: raw gfx1250 encoding via asm. VDST = LDS byte address (VGPR),
  // VADDR = 64-bit global address (VGPR pair).
  unsigned lds_addr =
      (unsigned)(unsigned long long)((AS3 char*)lds_generic + lds_byte_off);
  unsigned long long gaddr = (unsigned long long)gsrc;
  asm volatile("global_load_async_to_lds_b128 %0, %1, off"
               :
               : "v"(lds_addr), "v"(gaddr)
               : "memory");
#endif
}

__device__ __forceinline__ void wait_asynccnt0() {
#if defined(HAVE_WAIT_ASYNC)
  __builtin_amdgcn_s_wait_asynccnt(0);
#else
  asm volatile("s_wait_asynccnt 0" ::: "memory");
#endif
}

__global__ __launch_bounds__(256) void span_mean_pool_wmma(
    const float* __restrict__ emb,        // [B, L, D] f32
    const int* __restrict__ span_pos,     // [B, S, 2] i32 (start-1, end-1)
    const int* __restrict__ span_nums,    // [B] i32
    float* __restrict__ out_pooled,       // [B, S, D] f32
    float* __restrict__ out_mask)         // [B, S] (bool as f32)
{
  __shared__ float tile[MAX_ROWS * LDS_STRIDE];   // ~50 KB of 320 KB WGP LDS

  const int bs = blockIdx.x;          // span id: b*S + s
  const int b  = bs / Sc;
  const int s  = bs - b * Sc;
  const int t  = threadIdx.x;

  // sp = span_positions + 1  ->  start, end
  const int start = span_pos[bs * 2 + 0] + 1;
  const int end   = span_pos[bs * 2 + 1] + 1;
  int len = end - start;
  if (len < 1) len = 1;
  if (len > MAX_ROWS) len = MAX_ROWS;

  const float* gbase = emb + ((size_t)b * Lc + start) * Dc;

  // ---- Phase 1a: zero-fill LDS rows >= len (so weight-0 rows multiply 0*0).
  {
    float4 z = make_float4(0.f, 0.f, 0.f, 0.f);
    for (int q = t; q < MAX_ROWS * ROW_CHUNKS16; q += 256) {
      const int row = q / ROW_CHUNKS16;
      const int c16 = q - row * ROW_CHUNKS16;
      if (row >= len) {
        *(float4*)&tile[row * LDS_STRIDE + c16 * 4] = z;
      }
    }
  }

  // ---- Phase 1b: async-stage valid rows global -> LDS, 16B per lane per op.
  // 16 rows x 192 chunks = 3072 chunks over 256 threads = 12 iterations.
  {
#pragma unroll
    for (int i = 0; i < 12; ++i) {
      const int q   = t + i * 256;
      const int row = q / ROW_CHUNKS16;
      const int c16 = q - row * ROW_CHUNKS16;
      if (row < len) {
        async_copy_b128(gbase + (size_t)row * Dc + c16 * 4,
                        tile,
                        (unsigned)(row * LDS_STRIDE + c16 * 4) * 4u);
      }
    }
  }

  wait_asynccnt0();      // per-wave: our async copies landed in LDS
  __syncthreads();       // cross-wave visibility (also covers the DS zeroes)

  // ---- Phase 2: weighted reduction via chained V_WMMA_F32_16X16X4_F32.
  const int lane = t & 31;
  const int wave = t >> 5;              // 0..7
  const int half = (lane >> 4) & 1;     // which 16-lane half
  const int n    = lane & 15;           // N index within chunk

  const float inv   = 1.0f / (float)len;
  const float maskv = (s < span_nums[b]) ? 1.0f : 0.0f;

  float* outp = out_pooled + (size_t)bs * Dc;

  // 48 sixteen-column chunks of D=768; wave w owns chunks w, w+8, ...
#pragma unroll
  for (int i = 0; i < 6; ++i) {
    const int chunk = wave + 8 * i;     // uniform per wave
    const int col0  = chunk * 16;

    v8f c = {0.f, 0.f, 0.f, 0.f, 0.f, 0.f, 0.f, 0.f};

#pragma unroll
    for (int j = 0; j < 4; ++j) {
      // A (16x4 f32): VGPR0 = K{0|2}, VGPR1 = K{1|3} per half-wave.
      // All M rows share the weight vector -> broadcast per half.
      const int r0 = 4 * j + (half ? 2 : 0);
      v2f a, bm;
      a.x = (r0     < len) ? inv : 0.0f;
      a.y = (r0 + 1 < len) ? inv : 0.0f;
      // B (4x16 f32): lanes 0-15 = rows r0/r0+1, lanes 16-31 = rows r0+2/r0+3
      bm.x = tile[r0       * LDS_STRIDE + col0 + n];
      bm.y = tile[(r0 + 1) * LDS_STRIDE + col0 + n];
      c = __builtin_amdgcn_wmma_f32_16x16x4_f32(
          /*neg_a=*/false, a, /*neg_b=*/false, bm,
          /*c_mod=*/(short)0, c, /*reuse_a=*/false, /*reuse_b=*/false);
    }

    // All D rows are identical; row 0 lives in c[0] on lanes 0-15.
    if (lane < 16) {
      outp[col0 + lane] = c[0] * maskv;
    }
  }

  if (t == 0) {
    out_mask[bs] = maskv;
  }
}

extern "C" void kernel_launch(void* const* d_in, const int* in_sizes, int n_in,
                              void* d_out, int out_size, void* d_ws,
                              size_t ws_size, hipStream_t stream) {
  (void)in_sizes; (void)n_in; (void)out_size; (void)d_ws; (void)ws_size;

  const float* emb       = (const float*)d_in[0];  // embeddings   [B,L,D] f32
  const int*   span_pos  = (const int*)d_in[1];    // span_positions [B,S,2] i32
  const int*   span_nums = (const int*)d_in[2];    // span_nums    [B] i32
  // d_in[3] (repeated_vector) is derivable; unused.

  float* out        = (float*)d_out;
  float* out_pooled = out;                               // B*S*D floats
  float* out_mask   = out + (size_t)Bc * Sc * Dc;        // B*S mask values

  dim3 grid(Bc * Sc), block(256);
  hipLaunchKernelGGL(span_mean_pool_wmma, grid, block, 0, stream,
                     emb, span_pos, span_nums, out_pooled, out_mask);
}